// BilateralDenoiser_41910290874518
// MI455X (gfx1250) — compile-verified
//
#include <hip/hip_runtime.h>
#include <hip/hip_bf16.h>

// Bilateral denoiser, (2,3,224,224) fp32.
// Compute-bound stencil: 225 taps/pixel, ~68M v_exp_f32 + ~360M VALU total,
// only ~2.4MB of DRAM traffic -> LDS tiling with CDNA5 async global->LDS
// copies is the right MI455X path (no matmul structure, so no WMMA).
//
// Per-tap math is strength-reduced: log2(e) and the sigma scales are folded
// into per-row/per-column constants so each tap is 6 VALU + 1 v_exp_f32
// (TRANS, co-executes with VALU).

#define KF      15
#define PADR    7                 // KF/2
#define TILE    16
#define HALO    (TILE + KF - 1)   // 30
#define LSTRIDE 32                // LDS row stride (conflict-free for wave32)
#define HW      224

#if __has_builtin(__builtin_amdgcn_global_load_async_to_lds_b32) && \
    __has_builtin(__builtin_amdgcn_s_wait_asynccnt)
#define USE_ASYNC_LDS 1
#else
#define USE_ASYNC_LDS 0
#endif

#if USE_ASYNC_LDS
// Builtin params are int* in AS1 (global) and AS3 (LDS).
typedef __attribute__((address_space(1))) int as1_int;
typedef __attribute__((address_space(3))) int as3_int;
#endif

__device__ __forceinline__ float exp2_fast(float a) {
#if __has_builtin(__builtin_amdgcn_exp2f)
  return __builtin_amdgcn_exp2f(a);   // single v_exp_f32
#else
  return exp2f(a);
#endif
}

__global__ __launch_bounds__(256)
void bilateral_tile_kernel(const float* __restrict__ x,
                           const float* __restrict__ blur_sigma_p,
                           const float* __restrict__ diff_sigma_p,
                           float* __restrict__ out)
{
  __shared__ float tile[HALO * LSTRIDE];

  const int plane = blockIdx.z;                 // 0..5 (b*C + c)
  const int tx0   = blockIdx.x * TILE;          // 0..208, step 16
  const int ty0   = blockIdx.y * TILE;
  const float* __restrict__ src = x + (size_t)plane * (HW * HW);

  const float L2E     = 1.44269504088896340736f;
  const float bs      = blur_sigma_p[0];
  const float dsg     = diff_sigma_p[0];
  const float nb      = -L2E / (bs * bs);       // -log2(e)/blur_sigma^2
  const float nids2   = -L2E / (dsg * dsg);     // -log2(e)/diff_sigma^2

  const int tid = threadIdx.x;

  // ---- Stage halo tile (30x30 around the 16x16 output tile) into LDS ----
  // In-bounds cells: CDNA5 async global->LDS copy (ASYNCcnt).
  // Out-of-bounds cells: ds_store of the reference's -100 pad value
  // (disjoint LDS addresses -> no DS/ASYNC ordering hazard).
  for (int i = tid; i < HALO * HALO; i += 256) {
    const int ly = i / HALO;
    const int lxh = i - ly * HALO;
    const int gy = ty0 + ly - PADR;
    const int gx = tx0 + lxh - PADR;
    float* dst = &tile[ly * LSTRIDE + lxh];
    const bool inb = (gy >= 0) & (gy < HW) & (gx >= 0) & (gx < HW);
    if (inb) {
#if USE_ASYNC_LDS
      __builtin_amdgcn_global_load_async_to_lds_b32(
          (as1_int*)(src + (size_t)gy * HW + gx),
          (as3_int*)dst, /*imm offset=*/0, /*cpol=*/0);
#else
      *dst = src[(size_t)gy * HW + gx];
#endif
    } else {
      *dst = -100.0f;
    }
  }
#if USE_ASYNC_LDS
  __builtin_amdgcn_s_wait_asynccnt(0);   // drain ASYNCcnt before the barrier
#endif
  __syncthreads();

  // ---- Per-thread pixel ----
  const int lx = tid & (TILE - 1);      // 0..15
  const int ly = tid >> 4;              // 0..15
  const float* __restrict__ wrow = &tile[ly * LSTRIDE + lx];  // window origin
  const float c = wrow[PADR * LSTRIDE + PADR];                // center value

  // Pre-scaled spatial column terms: nbx[kx] = -(kx-7)^2 * log2e / bs^2
  float nbx[KF];
#pragma unroll
  for (int kx = 0; kx < KF; ++kx)
    nbx[kx] = (float)((kx - PADR) * (kx - PADR)) * nb;

  float num = 0.0f;   // sum p * e
  float den = 0.0f;   // sum e      (>= 1: center tap has exponent 0)

  for (int ky = 0; ky < KF; ++ky) {
    const float nby = (float)((ky - PADR) * (ky - PADR)) * nb;
    const float* __restrict__ row = wrow + ky * LSTRIDE;
#pragma unroll
    for (int kx = 0; kx < KF; ++kx) {
      const float p   = row[kx];
      const float d   = p - c;
      // arg = -log2e * ((p-c)^2/ds^2 + dist^2/bs^2)  <= 0, == 0 at center
      const float arg = fmaf(d * d, nids2, nby + nbx[kx]);
      const float e   = exp2_fast(arg);        // exp(-l); pad taps -> 0
      den += e;
      num = fmaf(p, e, num);
    }
  }

  float* __restrict__ dstp = out + (size_t)plane * (HW * HW);
  dstp[(size_t)(ty0 + ly) * HW + (tx0 + lx)] = num / den;
}

extern "C" void kernel_launch(void* const* d_in, const int* in_sizes, int n_in,
                              void* d_out, int out_size, void* d_ws, size_t ws_size,
                              hipStream_t stream) {
  (void)in_sizes; (void)n_in; (void)d_ws; (void)ws_size; (void)out_size;
  const float* x  = (const float*)d_in[0];
  const float* bs = (const float*)d_in[1];   // blur_sigma (1 elem)
  const float* ds = (const float*)d_in[2];   // diff_sigma (1 elem)
  // d_in[3] = filter_size (==15, hardcoded at compile time)
  float* out = (float*)d_out;

  dim3 grid(HW / TILE, HW / TILE, 6);  // 14 x 14 x (B*C)
  dim3 block(256);
  bilateral_tile_kernel<<<grid, block, 0, stream>>>(x, bs, ds, out);
}